// KGATEncoder_12429635354866
// MI455X (gfx1250) — compile-verified
//
#include <hip/hip_runtime.h>
#include <hip/hip_bf16.h>

// KGAT encoder for MI455X (gfx1250, wave32).
// SpMM scatter via L2-resident f32 atomics; dense GEMMs via V_WMMA_F32_16X16X4_F32
// (fp32-native WMMA — GEMM is not the bottleneck, so keep full precision).

typedef __attribute__((ext_vector_type(2))) float v2f;
typedef __attribute__((ext_vector_type(8))) float v8f;

#define LN_EPS 1e-5f
#define LRELU_SLOPE 0.01f
#define OUT_STRIDE 176  // 64 + 64 + 32 + 16

// ---------------------------------------------------------------- copy emb
__global__ void copy_emb_kernel(const float* __restrict__ emb,
                                float* __restrict__ out, int total) {
  int idx = blockIdx.x * blockDim.x + threadIdx.x;
  int stride = gridDim.x * blockDim.x;
  for (int i = idx; i < total; i += stride) {
    int node = i >> 6;   // d = 64
    int j = i & 63;
    out[(size_t)node * OUT_STRIDE + j] = emb[i];
  }
}

// ---------------------------------------------------------------- SpMM scatter
// side[rows[e], :] += vals[e] * ego[cols[e], :]
// One wave per edge per iteration: lanes cover the feature dimension
// (coalesced 128B gather from L2, global_atomic_add_f32 scatter).
template <int DIN>
__global__ void spmm_kernel(const int* __restrict__ rows,
                            const int* __restrict__ cols,
                            const float* __restrict__ vals,
                            const float* __restrict__ ego,
                            float* __restrict__ side, int nEdges) {
  int lane = threadIdx.x & 31;
  int wave = (blockIdx.x * blockDim.x + threadIdx.x) >> 5;
  int nWaves = (gridDim.x * blockDim.x) >> 5;
  for (int e = wave; e < nEdges; e += nWaves) {
    int c = cols[e];
    int r = rows[e];
    float v = vals[e];
#pragma unroll
    for (int j = 0; j < DIN / 32; ++j) {
      int f = lane + j * 32;
      float x = ego[(size_t)c * DIN + f];
      atomicAdd(&side[(size_t)r * DIN + f], v * x);
    }
  }
}

// ---------------------------------------------------------------- layer kernel
// One wave = one 16-node tile. Both GEMMs (sum path and product path) computed
// with V_WMMA_F32_16X16X4_F32; post-ops (bias, leaky-relu, LayerNorm, combine,
// L2-normalize) done through per-wave LDS staging.
//
// A 16x4 f32 fragment layout (ISA 7.12.2): lane L<16 holds X[M=L][k0], X[M=L][k0+1];
// lane L>=16 holds X[M=L-16][k0+2], X[M=L-16][k0+3].
// B 4x16 fragment: lane L<16 holds W[k0][N=L], W[k0+1][N=L]; lanes 16-31 hold K=2,3.
// C/D 16x16 f32: VGPR j, lanes 0-15 -> [M=j][N=lane], lanes 16-31 -> [M=j+8][N=lane-16].
template <int DIN, int DOUT>
__global__ void __launch_bounds__(128) layer_kernel(
    const float* __restrict__ ego, const float* __restrict__ side,
    const float* __restrict__ w1, const float* __restrict__ b1,
    const float* __restrict__ w2, const float* __restrict__ b2,
    const float* __restrict__ g1, const float* __restrict__ be1,
    const float* __restrict__ g2, const float* __restrict__ be2,
    float* __restrict__ ego_out, float* __restrict__ out, int outCol,
    int nNodes) {
  constexpr int NT = DOUT / 16;  // N-tiles of 16
  constexpr int LSTR = 65;       // padded LDS row stride (bank-conflict free)
  __shared__ float lds[4][2][16 * LSTR];

  const int wid = threadIdx.x >> 5;
  const int lane = threadIdx.x & 31;
  const int tile = blockIdx.x * 4 + wid;
  if (tile * 16 >= nNodes) return;  // wave-uniform: EXEC stays all-1s for WMMA
  const int row0 = tile * 16;

  const v8f vzero = {0.f, 0.f, 0.f, 0.f, 0.f, 0.f, 0.f, 0.f};
  v8f acc1[NT], acc2[NT];
#pragma unroll
  for (int t = 0; t < NT; ++t) { acc1[t] = vzero; acc2[t] = vzero; }

  int mrow = row0 + (lane & 15);
  if (mrow >= nNodes) mrow = nNodes - 1;  // clamp loads only; no EXEC divergence
  const int khalf = (lane >> 4) * 2;
  const int ncol0 = lane & 15;

  for (int k0 = 0; k0 < DIN; k0 += 4) {
    const float* pe = ego + (size_t)mrow * DIN + (k0 + khalf);
    const float* ps = side + (size_t)mrow * DIN + (k0 + khalf);
    float e0 = pe[0], e1 = pe[1];
    float s0 = ps[0], s1 = ps[1];
    v2f aS = {e0 + s0, e1 + s1};  // (ego + side) fragment
    v2f aP = {e0 * s0, e1 * s1};  // (ego * side) fragment
#pragma unroll
    for (int t = 0; t < NT; ++t) {
      int nc = t * 16 + ncol0;
      v2f bA = {w1[(size_t)(k0 + khalf) * DOUT + nc],
                w1[(size_t)(k0 + khalf + 1) * DOUT + nc]};
      v2f bB = {w2[(size_t)(k0 + khalf) * DOUT + nc],
                w2[(size_t)(k0 + khalf + 1) * DOUT + nc]};
      acc1[t] = __builtin_amdgcn_wmma_f32_16x16x4_f32(
          false, aS, false, bA, (short)0, acc1[t], false, false);
      acc2[t] = __builtin_amdgcn_wmma_f32_16x16x4_f32(
          false, aP, false, bB, (short)0, acc2[t], false, false);
    }
  }

  // Stage both 16xDOUT tiles in this wave's private LDS region.
  float* h1 = &lds[wid][0][0];
  float* h2 = &lds[wid][1][0];
  const int mbase = (lane >> 4) * 8;
#pragma unroll
  for (int t = 0; t < NT; ++t) {
#pragma unroll
    for (int j = 0; j < 8; ++j) {
      h1[(mbase + j) * LSTR + t * 16 + ncol0] = acc1[t][j];
      h2[(mbase + j) * LSTR + t * 16 + ncol0] = acc2[t][j];
    }
  }
  asm volatile("s_wait_dscnt 0x0" ::: "memory");  // same-wave LDS visibility

  // Phase 1: per-row bias + leaky-relu + LayerNorm, in place.
  // lane&15 = row; lanes 0-15 process h1 (s path), lanes 16-31 process h2 (bi path).
  {
    const int r = lane & 15;
    float* h = (lane < 16) ? h1 : h2;
    const float* bb = (lane < 16) ? b1 : b2;
    const float* gg = (lane < 16) ? g1 : g2;
    const float* bet = (lane < 16) ? be1 : be2;
    float m = 0.f;
#pragma unroll
    for (int f = 0; f < DOUT; ++f) {
      float x = h[r * LSTR + f] + bb[f];
      x = (x > 0.f) ? x : x * LRELU_SLOPE;
      m += x;
    }
    m *= (1.f / DOUT);
    float var = 0.f;
#pragma unroll
    for (int f = 0; f < DOUT; ++f) {
      float x = h[r * LSTR + f] + bb[f];
      x = (x > 0.f) ? x : x * LRELU_SLOPE;
      float d = x - m;
      var += d * d;
    }
    var *= (1.f / DOUT);
    float inv = rsqrtf(var + LN_EPS);
#pragma unroll
    for (int f = 0; f < DOUT; ++f) {
      float x = h[r * LSTR + f] + bb[f];
      x = (x > 0.f) ? x : x * LRELU_SLOPE;
      h[r * LSTR + f] = (x - m) * inv * gg[f] + bet[f];
    }
  }
  asm volatile("s_wait_dscnt 0x0" ::: "memory");

  // Phase 2: ego_new = s + bi (raw -> ego_out for next layer),
  // L2-normalized -> output slab. Lanes 0-15 own the 16 rows.
  if (lane < 16) {
    const int node = row0 + lane;
    if (node < nNodes) {
      const int r = lane;
      float ss = 0.f;
#pragma unroll
      for (int f = 0; f < DOUT; ++f) {
        float v = h1[r * LSTR + f] + h2[r * LSTR + f];
        ego_out[(size_t)node * DOUT + f] = v;
        ss += v * v;
      }
      float inv = 1.f / fmaxf(sqrtf(ss), 1e-12f);
#pragma unroll
      for (int f = 0; f < DOUT; ++f) {
        float v = h1[r * LSTR + f] + h2[r * LSTR + f];
        out[(size_t)node * OUT_STRIDE + outCol + f] = v * inv;
      }
    }
  }
}

// ---------------------------------------------------------------- launch
extern "C" void kernel_launch(void* const* d_in, const int* in_sizes, int n_in,
                              void* d_out, int out_size, void* d_ws,
                              size_t ws_size, hipStream_t stream) {
  const float* emb = (const float*)d_in[0];
  const int* rows = (const int*)d_in[1];
  const int* cols = (const int*)d_in[2];
  const float* vals = (const float*)d_in[3];
  const int N = in_sizes[0] / 64;  // 100000
  const int E = in_sizes[1];       // 3200000
  float* out = (float*)d_out;

  // Workspace: side (N*64), egoA (N*64), egoB (N*64) f32 -> ~76.8 MB.
  float* side = (float*)d_ws;
  float* egoA = side + (size_t)N * 64;
  float* egoB = egoA + (size_t)N * 64;

  const int nTiles = (N + 15) / 16;
  const int layerBlocks = (nTiles + 3) / 4;  // 4 waves (tiles) per block

  // out[:, 0:64] = emb (un-normalized, per reference)
  copy_emb_kernel<<<4096, 256, 0, stream>>>(emb, out, N * 64);

  // ---- layer 0: 64 -> 64
  hipMemsetAsync(side, 0, (size_t)N * 64 * sizeof(float), stream);
  spmm_kernel<64><<<2048, 256, 0, stream>>>(rows, cols, vals, emb, side, E);
  layer_kernel<64, 64><<<layerBlocks, 128, 0, stream>>>(
      emb, side, (const float*)d_in[4], (const float*)d_in[5],
      (const float*)d_in[6], (const float*)d_in[7], (const float*)d_in[8],
      (const float*)d_in[9], (const float*)d_in[10], (const float*)d_in[11],
      egoA, out, 64, N);

  // ---- layer 1: 64 -> 32
  hipMemsetAsync(side, 0, (size_t)N * 64 * sizeof(float), stream);
  spmm_kernel<64><<<2048, 256, 0, stream>>>(rows, cols, vals, egoA, side, E);
  layer_kernel<64, 32><<<layerBlocks, 128, 0, stream>>>(
      egoA, side, (const float*)d_in[12], (const float*)d_in[13],
      (const float*)d_in[14], (const float*)d_in[15], (const float*)d_in[16],
      (const float*)d_in[17], (const float*)d_in[18], (const float*)d_in[19],
      egoB, out, 128, N);

  // ---- layer 2: 32 -> 16
  hipMemsetAsync(side, 0, (size_t)N * 32 * sizeof(float), stream);
  spmm_kernel<32><<<2048, 256, 0, stream>>>(rows, cols, vals, egoB, side, E);
  layer_kernel<32, 16><<<layerBlocks, 128, 0, stream>>>(
      egoB, side, (const float*)d_in[20], (const float*)d_in[21],
      (const float*)d_in[22], (const float*)d_in[23], (const float*)d_in[24],
      (const float*)d_in[25], (const float*)d_in[26], (const float*)d_in[27],
      egoA, out, 160, N);
}